// DenseGATConv_42322607735060
// MI455X (gfx1250) — compile-verified
//
#include <hip/hip_runtime.h>

// ---------------------------------------------------------------------------
// DenseGATConv with use_diff=True (x replaced by ones) collapses analytically:
//   out[b,n,c] = (1/H) * sum_h sum_f W_lin[f, h*C + c]
// because xh is row-constant and softmax rows sum to 1 (self loops guarantee
// a nonempty row). So: (1) reduce W_lin columns with exact-f32 WMMA
// (V_WMMA_F32_16X16X4_F32, A = ones), (2) broadcast 64 floats into d_out.
// ---------------------------------------------------------------------------

typedef __attribute__((ext_vector_type(2))) float v2f;
typedef __attribute__((ext_vector_type(8))) float v8f;

#define F_DIM 128   // input features (K of the ones @ W_lin GEMM)
#define HC    256   // H*C columns of W_lin
#define C_DIM 64
#define NBLK2 512

// Kernel 1: one workgroup, 16 waves (wave32). Wave w owns columns
// [16w, 16w+16). Accumulate C += ones(16x4) x W[4x16] over K via f32 WMMA.
// A-matrix 16x4 f32 layout: lanes 0-15 hold K=0(V0),K=1(V1); lanes 16-31 hold
// K=2(V0),K=3(V1) -> ones everywhere. B-matrix mirrors with N across lanes.
// D-matrix: row M=0 sits in VGPR0 on lanes 0-15 (N = lane).
__global__ __launch_bounds__(512) void gat_colsum_wmma(
    const float* __restrict__ W,   // [F_DIM, HC] row-major
    float* __restrict__ y)         // [C_DIM] result in workspace
{
    __shared__ float cs[HC];

    const int tid  = threadIdx.x;
    const int wave = tid >> 5;           // 0..15
    const int lane = tid & 31;
    const int half = lane >> 4;          // selects K-pair {0,1} vs {2,3}
    const int n    = (wave << 4) + (lane & 15);  // column index 0..255

    v2f a; a.x = 1.0f; a.y = 1.0f;       // A = ones(16x4), all lanes
    v8f acc = {};

    #pragma unroll
    for (int f = 0; f < F_DIM; f += 4) {
        v2f b;
        b.x = W[(f + 2 * half    ) * HC + n];
        b.y = W[(f + 2 * half + 1) * HC + n];
        acc = __builtin_amdgcn_wmma_f32_16x16x4_f32(
            /*neg_a=*/false, a, /*neg_b=*/false, b,
            /*c_mod=*/(short)0, acc, /*reuse_a=*/false, /*reuse_b=*/false);
    }

    if (lane < 16) cs[n] = acc[0];       // M=0 row == column sum
    __syncthreads();

    if (tid < C_DIM) {
        y[tid] = 0.25f * (cs[tid] + cs[tid + 64] + cs[tid + 128] + cs[tid + 192]);
    }
}

// Kernel 2: broadcast y[0..63] over out[B*N*C] with 128-bit stores.
// idx*4 mod 64 == 4*(idx & 15), so each thread's float4 is y4[idx & 15].
__global__ __launch_bounds__(256) void gat_broadcast(
    const float* __restrict__ y, float4* __restrict__ out4, int total4)
{
    const float4* y4 = (const float4*)y;
    int idx = blockIdx.x * blockDim.x + threadIdx.x;
    const int stride = gridDim.x * blockDim.x;   // multiple of 16 -> pattern-safe
    for (; idx < total4; idx += stride) {
        out4[idx] = y4[idx & 15];
    }
}

extern "C" void kernel_launch(void* const* d_in, const int* in_sizes, int n_in,
                              void* d_out, int out_size, void* d_ws, size_t ws_size,
                              hipStream_t stream) {
    // setup_inputs order: x(0), adj(1), diff(2), W_lin(3), w_diff(4),
    //                     att_src(5), att_dst(6) -- only W_lin matters.
    const float* W_lin = (const float*)d_in[3];
    float* y   = (float*)d_ws;     // 64 floats of scratch
    float* out = (float*)d_out;    // [B*N*C] = 524288 floats

    gat_colsum_wmma<<<1, 512, 0, stream>>>(W_lin, y);

    const int total4 = out_size >> 2;            // 131072 float4 stores
    int blocks = (total4 + 255) / 256;
    if (blocks > NBLK2) blocks = NBLK2;
    gat_broadcast<<<blocks, 256, 0, stream>>>(y, (float4*)out, total4);
}